// CenterGitLoss_79894981640596
// MI455X (gfx1250) — compile-verified
//
#include <hip/hip_runtime.h>

// Problem constants (from reference): B=8192 rows, C=10000 centers, D=128 dims.
#define B_N 8192
#define C_N 10000
#define D_N 128
#define CD  (C_N * D_N)
#define CLAMP_MIN 1e-12f
#define CLAMP_MAX 1e12f
#define LDS_PITCH 132     // 128 + 4 pad DWORDs (TDM pad_interval=6, pad_amount=3) -> conflict-free ds_load_b64
#define TILE_FLOATS (16 * LDS_PITCH)
#define NTILES (C_N / 16)   // 625, exact

typedef __attribute__((ext_vector_type(2))) float v2f;
typedef __attribute__((ext_vector_type(8))) float v8f;
typedef __attribute__((ext_vector_type(4))) unsigned int v4u;
typedef __attribute__((ext_vector_type(8))) int v8i;
typedef __attribute__((ext_vector_type(4))) int v4i;

// ---------------- Stage 0: zero accumulators in workspace ----------------
__global__ void zero_ws_kernel(float* __restrict__ ws, float* __restrict__ acc) {
    int i = blockIdx.x * blockDim.x + threadIdx.x;
    if (i < CD + C_N) ws[i] = 0.0f;   // sum_x + counts
    if (i < 2) acc[i] = 0.0f;         // loss accumulators
}

// ---------------- Stage 1: counts, sum_x (scatter), x row norms ----------------
__global__ void scatter_stats_kernel(const float* __restrict__ x,
                                     const int* __restrict__ labels,
                                     float* __restrict__ sumx,
                                     float* __restrict__ counts,
                                     float* __restrict__ x2) {
    __shared__ float red[128];
    const int i = blockIdx.x;
    const int d = threadIdx.x;
    float v = x[i * D_N + d];
    red[d] = v * v;
    __syncthreads();
    for (int s = 64; s > 0; s >>= 1) {
        if (d < s) red[d] += red[d + s];
        __syncthreads();
    }
    if (d == 0) x2[i] = red[0];
    const int lab = labels[i];
    atomicAdd(&sumx[lab * D_N + d], v);
    if (d == 0) atomicAdd(&counts[lab], 1.0f);
}

// ---------------- Stage 2: center update (in-place over sum_x) + row norms ----------------
__global__ void update_centers_kernel(const float* __restrict__ centers,
                                      const float* __restrict__ counts,
                                      const float* __restrict__ lr,
                                      float* __restrict__ sumx_cnew,  // in: sum_x, out: centers_new
                                      float* __restrict__ cn2) {
    __shared__ float red[128];
    const int c = blockIdx.x;
    const int d = threadIdx.x;
    const float cnt = counts[c];
    const float cen = centers[c * D_N + d];
    const float s   = sumx_cnew[c * D_N + d];
    const float delta = (cnt * cen - s) / (1.0f + cnt);
    const float cn = cen - lr[0] * delta;
    sumx_cnew[c * D_N + d] = cn;
    red[d] = cn * cn;
    __syncthreads();
    for (int st = 64; st > 0; st >>= 1) {
        if (d < st) red[d] += red[d + st];
        __syncthreads();
    }
    if (d == 0) cn2[c] = red[0];
}

// ---------------- TDM: load one 16x128 f32 center tile into LDS with pitch-132 padding ----------------
// D# per CDNA5 ISA. 2D descriptor; groups 2/3 zero (tile_dim2=0 -> unused).
// amdgpu-toolchain (clang-23) 6-arg builtin form: (v4u, v8i, v4i, v4i, v8i, i32 cpol).
__device__ __forceinline__ void tdm_load_tile(unsigned int lds_addr, const float* gptr) {
    const unsigned long long ga = (unsigned long long)(uintptr_t)gptr;
    // group0: [1:0]=count=1 | [63:32]=lds_addr | [120:64]=global_addr | [127:126]=type 2
    v4u g0 = { 1u,
               lds_addr,
               (unsigned int)ga,
               (unsigned int)((ga >> 32) & 0x01FFFFFFu) | (2u << 30) };
    // group1 word0: data_size=2 (4B) <<16 | pad_enable<<20 | pad_interval=6 (128 DW) <<22 | pad_amount=3 (4 DW) <<25
    const unsigned int w0 = (2u << 16) | (1u << 20) | (6u << 22) | (3u << 25);
    v8i g1 = { (int)w0,
               (int)((D_N & 0xFFFFu) << 16),           // [79:48] tensor_dim0 = 128 (low 16 -> word1[31:16])
               (int)((16u & 0xFFFFu) << 16),           // tensor_dim0 hi16 = 0 | [111:80] tensor_dim1 = 16
               (int)((unsigned int)D_N << 16),         // tensor_dim1 hi16 = 0 | [127:112] tile_dim0 = 128
               (int)16,                                // [143:128] tile_dim1 = 16 | [159:144] tile_dim2 = 0
               (int)D_N,                               // [207:160] tensor_dim0_stride = 128 (low 32)
               0,                                      // stride0 hi16 | tensor_dim1_stride low16
               0 };                                    // tensor_dim1_stride hi
    v4i gz4 = { 0, 0, 0, 0 };
    v8i gz8 = { 0, 0, 0, 0, 0, 0, 0, 0 };
    __builtin_amdgcn_tensor_load_to_lds(g0, g1, gz4, gz4, gz8, 0);
}

// ---------------- Stage 3: fused distmat + losses via V_WMMA_F32_16X16X4_F32 ----------------
// 128 threads = 4 waves; wave w owns rows [blk*64 + 16w, +16). Column loop over 625 tiles of 16,
// center tiles double-buffered in LDS via the Tensor Data Mover (issued by wave 0, TENSORcnt-tracked).
__global__ void __launch_bounds__(128)
distloss_wmma_kernel(const float* __restrict__ x,
                     const int*   __restrict__ labels,
                     const float* __restrict__ cnew,
                     const float* __restrict__ cn2,
                     const float* __restrict__ x2,
                     float* __restrict__ acc) {
    __shared__ float btile[2 * TILE_FLOATS];

    const int tid  = threadIdx.x;
    const int wave = tid >> 5;
    const int lane = tid & 31;
    const int l16  = lane & 15;
    const int hi   = lane >> 4;

    const int row_base = blockIdx.x * 64 + wave * 16;

    // Preload A fragments: column-loop invariant. 32 K-steps of 4 -> 64 VGPRs/lane.
    const int am = row_base + l16;
    v2f a[32];
#pragma unroll
    for (int kk = 0; kk < 32; ++kk)
        a[kk] = *(const v2f*)&x[am * D_N + 4 * kk + 2 * hi];

    // Per-lane data for the C/D rows this lane owns (lanes 0-15: M=v, lanes 16-31: M=8+v).
    float x2r[8];
    int   labr[8];
#pragma unroll
    for (int v = 0; v < 8; ++v) {
        const int m = row_base + v + 8 * hi;
        x2r[v]  = x2[m];
        labr[v] = labels[m];
    }

    float gsum[8] = {0.f, 0.f, 0.f, 0.f, 0.f, 0.f, 0.f, 0.f};
    float cval[8] = {0.f, 0.f, 0.f, 0.f, 0.f, 0.f, 0.f, 0.f};
    float glab[8] = {0.f, 0.f, 0.f, 0.f, 0.f, 0.f, 0.f, 0.f};

    const unsigned int lds_base = (unsigned int)(uintptr_t)&btile[0];

    // Prologue: wave 0 DMAs tile 0 into buffer 0, waits TENSORcnt, barrier publishes it.
    if (wave == 0) {
        tdm_load_tile(lds_base, cnew);
        __builtin_amdgcn_s_wait_tensorcnt(0);
    }
    __syncthreads();

    int cur = 0;
    for (int t = 0; t < NTILES; ++t) {
        // Prefetch next tile into the other buffer while we compute (TDM overlaps WMMA).
        if (wave == 0 && (t + 1) < NTILES) {
            tdm_load_tile(lds_base + (unsigned int)((cur ^ 1) * TILE_FLOATS * 4),
                          cnew + (size_t)(t + 1) * 16 * D_N);
        }

        const float* bt = &btile[cur * TILE_FLOATS];
        v8f c = {};
#pragma unroll
        for (int kk = 0; kk < 32; ++kk) {
            v2f b = *(const v2f*)&bt[l16 * LDS_PITCH + 4 * kk + 2 * hi];
            c = __builtin_amdgcn_wmma_f32_16x16x4_f32(
                    /*neg_a=*/false, a[kk], /*neg_b=*/false, b,
                    /*c_mod=*/(short)0, c, /*reuse_a=*/false, /*reuse_b=*/false);
        }

        const int   col  = t * 16 + l16;
        const float cn2v = cn2[col];
#pragma unroll
        for (int v = 0; v < 8; ++v) {
            const float dist = x2r[v] + cn2v - 2.0f * c[v];
            const float git  = __builtin_amdgcn_rcpf(1.0f + dist);
            gsum[v] += git;
            if (labr[v] == col) { cval[v] += dist; glab[v] += git; }
        }

        if (wave == 0 && (t + 1) < NTILES) {
            __builtin_amdgcn_s_wait_tensorcnt(0);   // next tile landed in LDS
        }
        __syncthreads();                            // publish next buffer / retire current
        cur ^= 1;
    }

    // Reduce across the 16 lanes sharing each row set (masks 1,2,4,8 stay in-half).
#pragma unroll
    for (int v = 0; v < 8; ++v) {
#pragma unroll
        for (int m = 1; m <= 8; m <<= 1) {
            gsum[v] += __shfl_xor(gsum[v], m, 32);
            cval[v] += __shfl_xor(cval[v], m, 32);
            glab[v] += __shfl_xor(glab[v], m, 32);
        }
    }

    // Clip per row, sum the 8 rows each half owns.
    float cs = 0.f, gs = 0.f;
#pragma unroll
    for (int v = 0; v < 8; ++v) {
        cs += fminf(fmaxf(cval[v], CLAMP_MIN), CLAMP_MAX);
        gs += fminf(fmaxf(gsum[v] - glab[v], CLAMP_MIN), CLAMP_MAX);
    }
    const float cs16 = __shfl(cs, 16, 32);  // lane-16 half holds rows 8..15
    const float gs16 = __shfl(gs, 16, 32);
    if (lane == 0) {
        atomicAdd(&acc[0], cs + cs16);
        atomicAdd(&acc[1], gs + gs16);
    }
}

// ---------------- Stage 4: means ----------------
__global__ void finalize_kernel(const float* __restrict__ acc, float* __restrict__ out) {
    if (threadIdx.x == 0) {
        out[0] = acc[0] * (1.0f / (float)B_N);
        out[1] = acc[1] * (1.0f / (float)B_N);
    }
}

extern "C" void kernel_launch(void* const* d_in, const int* in_sizes, int n_in,
                              void* d_out, int out_size, void* d_ws, size_t ws_size,
                              hipStream_t stream) {
    const float* x       = (const float*)d_in[0];
    const int*   labels  = (const int*)d_in[1];
    const float* centers = (const float*)d_in[2];
    const float* lr      = (const float*)d_in[3];
    float* out = (float*)d_out;

    // Workspace layout (floats): sum_x[C*D] (reused as centers_new) | counts[C] | cn2[C] | x2[B] | acc[2]
    float* ws     = (float*)d_ws;
    float* sumx   = ws;
    float* counts = sumx + CD;
    float* cn2    = counts + C_N;
    float* x2     = cn2 + C_N;
    float* acc    = x2 + B_N;

    const int nzero = CD + C_N;
    zero_ws_kernel<<<(nzero + 255) / 256, 256, 0, stream>>>(ws, acc);
    scatter_stats_kernel<<<B_N, 128, 0, stream>>>(x, labels, sumx, counts, x2);
    update_centers_kernel<<<C_N, 128, 0, stream>>>(centers, counts, lr, sumx, cn2);
    distloss_wmma_kernel<<<B_N / 64, 128, 0, stream>>>(x, labels, sumx, cn2, x2, acc);
    finalize_kernel<<<1, 32, 0, stream>>>(acc, out);
}